// Gate_87479893885665
// MI455X (gfx1250) — compile-verified
//
#include <hip/hip_runtime.h>
#include <math.h>
#include <stdint.h>

#define DIMK      4096
#define NEXP      256
#define TOPKN     8
#define NGROUP    8
#define GSIZE     32
#define TOPKG     4
#define TOKENS    16384
#define SEQLEN    4096
#define NBATCH    4
#define WG_TOKENS 64           // tokens per workgroup (4 WMMA row-tiles)
#define CHUNK_K   256          // K staged per TDM transfer (64 KB)
#define NCHUNK    (DIMK / CHUNK_K)
#define NWG       (TOKENS / WG_TOKENS)      // 256
#define WGS_PER_B (SEQLEN / WG_TOKENS)      // 64
#define LDS_STR   260          // 256 + 4 pad
#define NEG_BIG   (-3.402823466e38f)
#define ROUTE_SCALE 2.5f
#define AUX_W     0.001f

typedef float v2f __attribute__((ext_vector_type(2)));
typedef float v8f __attribute__((ext_vector_type(8)));
typedef unsigned int v4u __attribute__((ext_vector_type(4)));
typedef int v8i __attribute__((ext_vector_type(8)));
typedef int v4i __attribute__((ext_vector_type(4)));

__device__ __forceinline__ float sigmoidf_fast(float v) {
    return 1.0f / (1.0f + __expf(-v));
}

// Issue one TDM transfer: 64 rows x CHUNK_K f32 tile of x -> LDS (contiguous).
__device__ __forceinline__ void tdm_load_chunk(unsigned lds_byte,
                                               unsigned long long gbyte) {
    v4u g0;
    g0.x = 1u;                                          // count=1, user mode
    g0.y = lds_byte;                                    // lds_addr
    g0.z = (unsigned)(gbyte & 0xFFFFFFFFull);           // global_addr[31:0]
    g0.w = (unsigned)((gbyte >> 32) & 0x1FFFFFFull)     // global_addr[56:32]
         | (2u << 30);                                  // type=2 ("image")
    v8i g1;
    g1[0] = (int)(2u << 16);                            // wg_mask=0, data_size=4B
    g1[1] = (int)(((unsigned)DIMK & 0xFFFFu) << 16);    // tensor_dim0[15:0]
    g1[2] = (int)(((unsigned)DIMK >> 16)                // tensor_dim0[31:16]
         | (((unsigned)TOKENS & 0xFFFFu) << 16));       // tensor_dim1[15:0]
    g1[3] = (int)(((unsigned)TOKENS >> 16)              // tensor_dim1[31:16]
         | ((unsigned)CHUNK_K << 16));                  // tile_dim0
    g1[4] = WG_TOKENS;                                  // tile_dim1 (tile_dim2=0)
    g1[5] = DIMK;                                       // tensor_dim0_stride[31:0]
    g1[6] = 0;                                          // stride0 hi | stride1 lo
    g1[7] = 0;
    v4i g2 = {0, 0, 0, 0};
    v4i g3 = {0, 0, 0, 0};
    v8i g4 = {0, 0, 0, 0, 0, 0, 0, 0};
    __builtin_amdgcn_tensor_load_to_lds(g0, g1, g2, g3, g4, 0);
}

__global__ __launch_bounds__(256) void gate_kernel(
    const float* __restrict__ x,        // [TOKENS, DIMK]
    const float* __restrict__ w,        // [NEXP, DIMK]
    const float* __restrict__ bias,     // [NEXP]
    float* __restrict__ out_w,          // [TOKENS, TOPKN]
    int*   __restrict__ out_idx,        // [TOKENS, TOPKN]
    float* __restrict__ partials)       // [NWG, 512]
{
    union SMem {
        float stage[2][WG_TOKENS * CHUNK_K];            // 2 x 64 KB (A tiles)
        struct {
            float scores[WG_TOKENS * LDS_STR];          // sigmoid scores
            float biasv[NEXP];
            int   sel[WG_TOKENS * TOPKN];
        } r;
    };
    __shared__ SMem sm;

    const int wg      = blockIdx.x;                     // 0..255
    const int tokBase = wg * WG_TOKENS;
    const int tid     = threadIdx.x;
    const int wave    = tid >> 5;                       // 0..7
    const int lane    = tid & 31;
    const int half    = lane >> 4;
    const int l15     = lane & 15;
    const int e0      = wave * 32;                      // 32 experts per wave

    const float* bPtr0 = w + (size_t)(e0 + l15) * DIMK + half * 2;
    const float* bPtr1 = bPtr0 + (size_t)16 * DIMK;

    v8f acc[8];                                         // [token-tile 0..3][exp-tile 0..1]
#pragma unroll
    for (int i = 0; i < 8; ++i) acc[i] = (v8f){0.f,0.f,0.f,0.f,0.f,0.f,0.f,0.f};

    const unsigned lds0      = (unsigned)(uintptr_t)(&sm.stage[0][0]);
    const unsigned ldsStride = (unsigned)(WG_TOKENS * CHUNK_K * 4);   // 64 KB
    const unsigned long long xByte = (unsigned long long)(uintptr_t)x
                                   + (unsigned long long)tokBase * DIMK * 4ull;

    // prologue: stage chunk 0
    if (tid < 32) tdm_load_chunk(lds0, xByte);

    for (int c = 0; c < NCHUNK; ++c) {
        __syncthreads();        // buf[(c+1)&1] free (chunk c-1 fully consumed)
        if (tid < 32) {
            if (c + 1 < NCHUNK) {
                tdm_load_chunk(lds0 + (unsigned)((c + 1) & 1) * ldsStride,
                               xByte + (unsigned long long)(c + 1) * CHUNK_K * 4ull);
                __builtin_amdgcn_s_wait_tensorcnt(1);   // chunk c complete
            } else {
                __builtin_amdgcn_s_wait_tensorcnt(0);
            }
        }
        __syncthreads();        // buf[c&1] ready for all waves

        const float* sb = sm.stage[c & 1];
        const int kg = c * CHUNK_K;
#pragma unroll 2
        for (int kk = 0; kk < CHUNK_K; kk += 4) {
            v2f a0 = *(const v2f*)(sb + (0 * 16 + l15) * CHUNK_K + kk + half * 2);
            v2f a1 = *(const v2f*)(sb + (1 * 16 + l15) * CHUNK_K + kk + half * 2);
            v2f a2 = *(const v2f*)(sb + (2 * 16 + l15) * CHUNK_K + kk + half * 2);
            v2f a3 = *(const v2f*)(sb + (3 * 16 + l15) * CHUNK_K + kk + half * 2);
            v2f b0 = *(const v2f*)(bPtr0 + kg + kk);
            v2f b1 = *(const v2f*)(bPtr1 + kg + kk);
            acc[0] = __builtin_amdgcn_wmma_f32_16x16x4_f32(false, a0, false, b0, (short)0, acc[0], false, false);
            acc[1] = __builtin_amdgcn_wmma_f32_16x16x4_f32(false, a0, false, b1, (short)0, acc[1], false, false);
            acc[2] = __builtin_amdgcn_wmma_f32_16x16x4_f32(false, a1, false, b0, (short)0, acc[2], false, false);
            acc[3] = __builtin_amdgcn_wmma_f32_16x16x4_f32(false, a1, false, b1, (short)0, acc[3], false, false);
            acc[4] = __builtin_amdgcn_wmma_f32_16x16x4_f32(false, a2, false, b0, (short)0, acc[4], false, false);
            acc[5] = __builtin_amdgcn_wmma_f32_16x16x4_f32(false, a2, false, b1, (short)0, acc[5], false, false);
            acc[6] = __builtin_amdgcn_wmma_f32_16x16x4_f32(false, a3, false, b0, (short)0, acc[6], false, false);
            acc[7] = __builtin_amdgcn_wmma_f32_16x16x4_f32(false, a3, false, b1, (short)0, acc[7], false, false);
        }
    }
    __syncthreads();            // done reading stage buffers -> safe to overlay

    // ---- sigmoid + stage scores into LDS; load bias vector ----
    if (tid < NEXP) sm.r.biasv[tid] = bias[tid];
#pragma unroll
    for (int tt = 0; tt < 4; ++tt)
#pragma unroll
        for (int n = 0; n < 2; ++n)
#pragma unroll
            for (int r = 0; r < 8; ++r) {
                const int t = tt * 16 + r + half * 8;
                const int e = e0 + n * 16 + l15;
                sm.r.scores[t * LDS_STR + e] = sigmoidf_fast(acc[tt * 2 + n][r]);
            }
    __syncthreads();

    // ---- routing: one thread per token (threads 0..63) ----
    if (tid < WG_TOKENS) {
        const float* srow = sm.r.scores + tid * LDS_STR;
        unsigned selm[8] = {0u,0u,0u,0u,0u,0u,0u,0u};

        float gs[NGROUP];
#pragma unroll
        for (int g = 0; g < NGROUP; ++g) {
            float m = NEG_BIG;
            for (int j = 0; j < GSIZE; ++j) {
                const int e = g * GSIZE + j;
                m = fmaxf(m, srow[e] + sm.r.biasv[e]);
            }
            gs[g] = m;
        }
        unsigned gmask = 0u;
#pragma unroll
        for (int it = 0; it < TOPKG; ++it) {
            float best = NEG_BIG; int bi = 0;
            for (int g = 0; g < NGROUP; ++g)
                if (gs[g] > best) { best = gs[g]; bi = g; }
            gmask |= 1u << bi;
            gs[bi] = NEG_BIG;
        }
        const int token = tokBase + tid;
        float selw[TOPKN]; int seli[TOPKN]; float wsum = 0.f;
        for (int it = 0; it < TOPKN; ++it) {
            float best = NEG_BIG; int bi = 0;
            for (int e = 0; e < NEXP; ++e) {
                if (((gmask >> (e >> 5)) & 1u) && !((selm[e >> 5] >> (e & 31)) & 1u)) {
                    const float v = srow[e] + sm.r.biasv[e];
                    if (v > best) { best = v; bi = e; }
                }
            }
            selm[bi >> 5] |= 1u << (bi & 31);
            seli[it] = bi;
            selw[it] = srow[bi];                 // raw sigmoid score
            wsum    += srow[bi];
            sm.r.sel[tid * TOPKN + it] = bi;
        }
        const float inv = ROUTE_SCALE / wsum;
#pragma unroll
        for (int it = 0; it < TOPKN; ++it) {
            out_w[token * TOPKN + it]   = selw[it] * inv;
            out_idx[token * TOPKN + it] = seli[it];
        }
    }
    __syncthreads();

    // ---- per-WG partials (fixed-order -> deterministic aux loss) ----
    {
        const int e = tid;                       // 0..255
        float ssum = 0.f;
        for (int t = 0; t < WG_TOKENS; ++t) ssum += sm.r.scores[t * LDS_STR + e];
        int cnt = 0;
        for (int t = 0; t < WG_TOKENS; ++t)
#pragma unroll
            for (int k2 = 0; k2 < TOPKN; ++k2) cnt += (sm.r.sel[t * TOPKN + k2] == e);
        partials[(size_t)wg * 512 + e]       = ssum;
        partials[(size_t)wg * 512 + 256 + e] = (float)cnt;
    }
}

__global__ __launch_bounds__(256) void aux_reduce_kernel(
    const float* __restrict__ partials, float* __restrict__ aux_out)
{
    __shared__ float red[256];
    const int e = threadIdx.x;                   // expert
    float acc = 0.f;
    for (int b = 0; b < NBATCH; ++b) {
        float ssum = 0.f, cnt = 0.f;
        for (int i = 0; i < WGS_PER_B; ++i) {
            const float* p = partials + (size_t)(b * WGS_PER_B + i) * 512;
            ssum += p[e];
            cnt  += p[256 + e];
        }
        acc += (cnt * (1.0f / 128.0f)) * (ssum * (1.0f / (float)SEQLEN));
    }
    red[e] = acc;
    __syncthreads();
    for (int s = 128; s > 0; s >>= 1) {
        if (e < s) red[e] += red[e + s];
        __syncthreads();
    }
    if (e == 0) *aux_out = red[0] * (1.0f / (float)NBATCH) * AUX_W;
}

extern "C" void kernel_launch(void* const* d_in, const int* in_sizes, int n_in,
                              void* d_out, int out_size, void* d_ws, size_t ws_size,
                              hipStream_t stream) {
    const float* x      = (const float*)d_in[0];   // [4,4096,4096] f32
    const float* weight = (const float*)d_in[1];   // [256,4096]    f32
    const float* bias   = (const float*)d_in[2];   // [256]         f32

    float* out_w   = (float*)d_out;                          // T*8 f32
    int*   out_idx = (int*)((float*)d_out + TOKENS * TOPKN); // T*8 i32
    float* aux     = (float*)d_out + 2 * TOKENS * TOPKN;     // 1   f32

    float* partials = (float*)d_ws;                // 256 * 512 f32 = 512 KB

    gate_kernel<<<NWG, 256, 0, stream>>>(x, weight, bias, out_w, out_idx, partials);
    aux_reduce_kernel<<<1, 256, 0, stream>>>(partials, aux);
}